// FNO2d_64957085384950
// MI455X (gfx1250) — compile-verified
//
#include <hip/hip_runtime.h>

typedef __attribute__((ext_vector_type(2))) float v2f;
typedef __attribute__((ext_vector_type(8))) float v8f;

#define DEV static __device__ __forceinline__

union V8U { v8f v; float f[8]; };

DEV v8f wmma4(v2f a, v2f b, v8f c) {
  // D = A(16x4) * B(4x16) + C, fp32
  return __builtin_amdgcn_wmma_f32_16x16x4_f32(false, a, false, b, (short)0, c, false, false);
}

DEV float gelu_tanh(float x) {
  const float c0 = 0.7978845608028654f, c1 = 0.044715f;
  float x3 = x * x * x;
  float t = tanhf(c0 * (x + c1 * x3));
  return 0.5f * x * (1.0f + t);
}

// ---------------- basis matrices (transposed [N][K] for v2f B-fragment loads) ---
// FyT  [48][128] : forward y-rDFT, cols padded 40->48
// Fx2T [80][256] : forward x-DFT on complex pairs (K = 2x+delta)
// E2   [128][80] : inverse x-DFT A-matrix (row = x, K = 2m+delta), scaled 1/128
// GyT  [128][40] : inverse y-rDFT (row = y, K = 2ky+delta), scaled 1/128 w/ 2x for ky>0
__global__ void k_bases(float* __restrict__ FyT, float* __restrict__ Fx2T,
                        float* __restrict__ E2, float* __restrict__ GyT) {
  int id = blockIdx.x * 256 + threadIdx.x;     // 41984 total
  const float W = 6.283185307179586f / 128.0f;
  int t = id;
  if (t < 6144) {                              // FyT
    int y = t & 127, n = t >> 7;
    float v = 0.0f;
    if (n < 40) {
      int ky = n >> 1;
      float th = W * (float)((ky * y) & 127);
      v = (n & 1) ? -sinf(th) : cosf(th);
    }
    FyT[t] = v;
    return;
  }
  t -= 6144;
  if (t < 20480) {                             // Fx2T
    int j = t & 255, n = t >> 8;
    int xx = j >> 1, d = j & 1;
    int m = n >> 1, c2 = n & 1;
    int kx = (m < 20) ? m : (108 + (m - 20));
    float th = W * (float)((kx * xx) & 127);
    float cs = cosf(th), sn = sinf(th);
    Fx2T[t] = (c2 == 0) ? (d ? sn : cs) : (d ? cs : -sn);
    return;
  }
  t -= 20480;
  if (t < 10240) {                             // E2
    int n = t % 80, xx = t / 80;
    int m = n >> 1, d = n & 1;
    int kx = (m < 20) ? m : (108 + (m - 20));
    float th = W * (float)((kx * xx) & 127);
    E2[t] = (d ? sinf(th) : cosf(th)) * (1.0f / 128.0f);
    return;
  }
  t -= 10240;                                  // GyT
  int j = t % 40, y = t / 40;
  int ky = j >> 1, d = j & 1;
  float s = ((ky == 0) ? 1.0f : 2.0f) / 128.0f;
  float th = W * (float)((ky * y) & 127);
  GyT[t] = d ? (-s * sinf(th)) : (s * cosf(th));
}

// wprep[m][ky][comp][o][i] = scw{1,2}[layer][i][o][m%20][ky][comp]  (BT layout)
__global__ void k_prepw(const float* __restrict__ scw1, const float* __restrict__ scw2,
                        float* __restrict__ wprep, int layer) {
  size_t id = (size_t)blockIdx.x * 256 + threadIdx.x;   // 6,553,600
  int i = (int)(id & 63); size_t t = id >> 6;
  int o = (int)(t & 63); t >>= 6;
  int comp = (int)(t & 1); t >>= 1;
  int ky = (int)(t % 20); int m = (int)(t / 20);
  const float* src = (m < 20) ? scw1 : scw2;
  int m1 = (m < 20) ? m : (m - 20);
  wprep[id] = src[(((((size_t)layer * 64 + i) * 64 + o) * 20 + m1) * 20 + ky) * 2 + comp];
}

// lift: H0[b][c][x][y] = concat(x, gridx, gridy) @ fc0_w + fc0_b
__global__ void k_fc0(const float* __restrict__ xin, const float* __restrict__ mx,
                      const float* __restrict__ my, const float* __restrict__ w,
                      const float* __restrict__ bias, float* __restrict__ H) {
  size_t id = (size_t)blockIdx.x * 256 + threadIdx.x;   // 16,777,216
  int y = (int)(id & 127);
  int xx = (int)((id >> 7) & 127);
  int c = (int)((id >> 14) & 63);
  int b = (int)(id >> 20);
  const float* p = xin + (((size_t)b * 128 + xx) * 128 + y) * 20;
  float acc = bias[c];
#pragma unroll
  for (int i = 0; i < 20; ++i) acc += p[i] * w[i * 64 + c];
  acc += mx[xx] * w[20 * 64 + c];
  acc += my[y] * w[21 * 64 + c];
  H[id] = acc;
}

// 1x1 conv (raw, no bias): Hout[b][o][s] = sum_i Hin[b][i][s] * ww[l][o][i]
__global__ void k_conv(const float* __restrict__ Hin, float* __restrict__ Hout,
                       const float* __restrict__ ww, int layer) {
  __shared__ float sB[64 * 64];                // BT[o][i] == ww[l] row-major
  for (int t = threadIdx.x; t < 4096; t += 256) sB[t] = ww[layer * 4096 + t];
  __syncthreads();
  int wid = (blockIdx.x * 256 + threadIdx.x) >> 5;   // 0..16383 row tiles
  int lane = threadIdx.x & 31;
  int b = wid >> 10;
  int s = (wid & 1023) * 16;
  int r = lane & 15, kb = (lane >> 4) << 1, radd = (lane >> 4) * 8;
  const float* Ab = Hin + ((size_t)b * 64) * 16384 + s + r;
  float areg[32];
#pragma unroll
  for (int t = 0; t < 16; ++t) {
    areg[2 * t]     = Ab[(size_t)(4 * t + kb) * 16384];
    areg[2 * t + 1] = Ab[(size_t)(4 * t + kb + 1) * 16384];
  }
  for (int nt = 0; nt < 4; ++nt) {
    int n = nt * 16 + (lane & 15);
    v8f acc = {};
#pragma unroll
    for (int t = 0; t < 16; ++t) {
      v2f a; a.x = areg[2 * t]; a.y = areg[2 * t + 1];
      v2f bb = *(const v2f*)(&sB[n * 64 + 4 * t + kb]);
      acc = wmma4(a, bb, acc);
    }
    V8U u; u.v = acc;
#pragma unroll
    for (int v = 0; v < 8; ++v)
      Hout[((size_t)b * 64 + n) * 16384 + s + v + radd] = u.f[v];
  }
}

// forward y-rDFT: yhatT[bc][ky][2x+d] = sum_y Hin[bc][x][y] * Fy[y][2ky+d]
__global__ void k_dfty(const float* __restrict__ Hin, float* __restrict__ yhatT,
                       const float* __restrict__ FyT) {
  __shared__ float sB[48 * 128];
  for (int t = threadIdx.x; t < 48 * 128; t += 256) sB[t] = FyT[t];
  __syncthreads();
  int wid = (blockIdx.x * 256 + threadIdx.x) >> 5;   // 0..8191
  int lane = threadIdx.x & 31;
  int bc = wid >> 3;
  int x0 = (wid & 7) * 16;
  int r = lane & 15, kb = (lane >> 4) << 1, radd = (lane >> 4) * 8;
  const float* Arow = Hin + ((size_t)bc * 128 + x0 + r) * 128;
  v2f areg[32];
#pragma unroll
  for (int t = 0; t < 32; ++t) areg[t] = *(const v2f*)(Arow + 4 * t + kb);
  for (int nt = 0; nt < 3; ++nt) {
    int n = nt * 16 + (lane & 15);
    v8f acc = {};
#pragma unroll
    for (int t = 0; t < 32; ++t) {
      v2f bb = *(const v2f*)(&sB[n * 128 + 4 * t + kb]);
      acc = wmma4(areg[t], bb, acc);
    }
    if (n < 40) {
      V8U u; u.v = acc;
      int ky = n >> 1, d = n & 1;
#pragma unroll
      for (int v = 0; v < 8; ++v) {
        int xx = x0 + v + radd;
        yhatT[(size_t)bc * 8192 + ky * 256 + 2 * xx + d] = u.f[v];
      }
    }
  }
}

// forward x-DFT: xft[m][ky][comp][b][i] = sum_{2x+d} yhatT[bc][ky][2x+d]*Fx2[2x+d][2m+comp]
__global__ void k_dftx(const float* __restrict__ yhatT, float* __restrict__ xft,
                       const float* __restrict__ Fx2T) {
  int wid = (blockIdx.x * 256 + threadIdx.x) >> 5;   // 0..10239
  int lane = threadIdx.x & 31;
  int t10 = wid % 10; int bc = wid / 10;
  int rowTile = t10 / 5, colTile = t10 % 5;
  int r = lane & 15, kb = (lane >> 4) << 1, n = colTile * 16 + (lane & 15);
  int radd = (lane >> 4) * 8;
  const float* Arow = yhatT + (size_t)bc * 8192 + (rowTile * 16 + r) * 256;
  v8f acc = {};
  for (int k0 = 0; k0 < 256; k0 += 4) {
    v2f a = *(const v2f*)(Arow + k0 + kb);
    v2f bb = *(const v2f*)(Fx2T + n * 256 + k0 + kb);
    acc = wmma4(a, bb, acc);
  }
  V8U u; u.v = acc;
  int b = bc >> 6, i = bc & 63;
  int m = n >> 1, comp = n & 1;
#pragma unroll
  for (int v = 0; v < 8; ++v) {
    int ky = rowTile * 16 + v + radd;
    if (ky < 20)
      xft[((((size_t)m * 20 + ky) * 2 + comp) * 16 + b) * 64 + i] = u.f[v];
  }
}

// per-(m,ky) complex channel mix, writes duplicated O2 (BT layout) for inverse x-DFT
__global__ void k_cmix(const float* __restrict__ xft, const float* __restrict__ wprep,
                       float* __restrict__ oft2) {
  int wid = (blockIdx.x * 256 + threadIdx.x) >> 5;   // 0..3199
  int lane = threadIdx.x & 31;
  int otile = wid & 3;
  int mk = wid >> 2;
  int ky = mk % 20, m = mk / 20;
  const float* AR = xft + (((size_t)m * 20 + ky) * 2 + 0) * 1024;
  const float* AI = AR + 1024;
  const float* BR = wprep + (((size_t)m * 20 + ky) * 2 + 0) * 4096;
  const float* BI = BR + 4096;
  int r = lane & 15, kb = (lane >> 4) << 1, nn = otile * 16 + (lane & 15);
  int radd = (lane >> 4) * 8;
  v8f cr = {}, ci = {};
  for (int k0 = 0; k0 < 64; k0 += 4) {
    v2f ar = *(const v2f*)(AR + r * 64 + k0 + kb);
    v2f ai = *(const v2f*)(AI + r * 64 + k0 + kb);
    v2f br = *(const v2f*)(BR + nn * 64 + k0 + kb);
    v2f bi = *(const v2f*)(BI + nn * 64 + k0 + kb);
    v2f nbi; nbi.x = -bi.x; nbi.y = -bi.y;
    cr = wmma4(ar, br, cr);
    cr = wmma4(ai, nbi, cr);
    ci = wmma4(ar, bi, ci);
    ci = wmma4(ai, br, ci);
  }
  V8U ur, ui; ur.v = cr; ui.v = ci;
#pragma unroll
  for (int v = 0; v < 8; ++v) {
    int b = v + radd;
    size_t base = (((size_t)b * 64 + nn) * 48 + 2 * ky) * 80 + 2 * m;
    oft2[base]          = ur.f[v];    // B[2m][2ky]   = oR
    oft2[base + 1]      = -ui.f[v];   // B[2m+1][2ky] = -oI
    oft2[base + 80]     = ui.f[v];    // B[2m][2ky+1] = oI
    oft2[base + 80 + 1] = ur.f[v];    // B[2m+1][2ky+1] = oR
  }
}

// inverse x-DFT: z[bo][x][2ky+c] = sum_{2m+d} E2[x][2m+d] * O2[bo][2ky+c][2m+d]
__global__ void k_idftx(const float* __restrict__ oft2, float* __restrict__ z,
                        const float* __restrict__ E2) {
  __shared__ float sA[128 * 80];
  for (int t = threadIdx.x; t < 128 * 80; t += 256) sA[t] = E2[t];
  __syncthreads();
  int wid = (blockIdx.x * 256 + threadIdx.x) >> 5;   // 0..24575
  int lane = threadIdx.x & 31;
  int t24 = wid % 24; int bo = wid / 24;
  int rowTile = t24 / 3, colTile = t24 % 3;
  int r = lane & 15, kb = (lane >> 4) << 1, n = colTile * 16 + (lane & 15);
  int radd = (lane >> 4) * 8;
  const float* Arow = sA + (rowTile * 16 + r) * 80;
  const float* Bb = oft2 + (size_t)bo * (48 * 80);
  v8f acc = {};
  for (int k0 = 0; k0 < 80; k0 += 4) {
    v2f a = *(const v2f*)(Arow + k0 + kb);
    v2f bb = *(const v2f*)(Bb + n * 80 + k0 + kb);
    acc = wmma4(a, bb, acc);
  }
  if (n < 40) {
    V8U u; u.v = acc;
#pragma unroll
    for (int v = 0; v < 8; ++v) {
      int x = rowTile * 16 + v + radd;
      z[((size_t)bo * 128 + x) * 40 + n] = u.f[v];
    }
  }
}

// inverse y-rDFT fused with skip-add + bias + gelu:
// Hout[bc][x][y] = act( sum_j z[bc][x][j]*Gy[j][y] + Hout(conv)[bc][x][y] + wb[l][c] )
__global__ void k_idfty(const float* __restrict__ z, float* __restrict__ Hout,
                        const float* __restrict__ GyT, const float* __restrict__ wb,
                        int layer) {
  __shared__ float sB[128 * 40];
  for (int t = threadIdx.x; t < 128 * 40; t += 256) sB[t] = GyT[t];
  __syncthreads();
  int wid = (blockIdx.x * 256 + threadIdx.x) >> 5;   // 0..8191
  int lane = threadIdx.x & 31;
  int bc = wid >> 3;
  int x0 = (wid & 7) * 16;
  int r = lane & 15, kb = (lane >> 4) << 1, radd = (lane >> 4) * 8;
  const float* Arow = z + ((size_t)bc * 128 + x0 + r) * 40;
  v2f areg[10];
#pragma unroll
  for (int t = 0; t < 10; ++t) areg[t] = *(const v2f*)(Arow + 4 * t + kb);
  float bias = wb[layer * 64 + (bc & 63)];
  for (int nt = 0; nt < 8; ++nt) {
    int n = nt * 16 + (lane & 15);
    v8f acc = {};
#pragma unroll
    for (int t = 0; t < 10; ++t) {
      v2f bb = *(const v2f*)(&sB[n * 40 + 4 * t + kb]);
      acc = wmma4(areg[t], bb, acc);
    }
    V8U u; u.v = acc;
#pragma unroll
    for (int v = 0; v < 8; ++v) {
      size_t idx = ((size_t)bc * 128 + x0 + v + radd) * 128 + n;
      float val = u.f[v] + Hout[idx] + bias;
      if (layer < 3) val = gelu_tanh(val);
      Hout[idx] = val;
    }
  }
}

// head: out[p] = fc2_b + sum_j gelu( sum_i H[b][i][p]*fc1_w[i][j] + fc1_b[j] ) * fc2_w[j]
__global__ void k_head(const float* __restrict__ Hin, const float* __restrict__ fc1w,
                       const float* __restrict__ fc1b, const float* __restrict__ fc2w,
                       const float* __restrict__ fc2b, float* __restrict__ out) {
  __shared__ float sBT[128 * 64];                 // fc1_w transposed [j][i]
  for (int t = threadIdx.x; t < 8192; t += 256) {
    int j = t >> 6, i = t & 63;
    sBT[t] = fc1w[i * 128 + j];
  }
  __syncthreads();
  int wid = (blockIdx.x * 256 + threadIdx.x) >> 5;   // 0..16383
  int lane = threadIdx.x & 31;
  int b = wid >> 10;
  int s = (wid & 1023) * 16;
  int r = lane & 15, kb = (lane >> 4) << 1;
  const float* Ab = Hin + ((size_t)b * 64) * 16384 + s + r;
  float areg[32];
#pragma unroll
  for (int t = 0; t < 16; ++t) {
    areg[2 * t]     = Ab[(size_t)(4 * t + kb) * 16384];
    areg[2 * t + 1] = Ab[(size_t)(4 * t + kb + 1) * 16384];
  }
  float psum[8];
#pragma unroll
  for (int v = 0; v < 8; ++v) psum[v] = 0.0f;
  for (int nt = 0; nt < 8; ++nt) {
    int n = nt * 16 + (lane & 15);
    v8f acc = {};
#pragma unroll
    for (int t = 0; t < 16; ++t) {
      v2f a; a.x = areg[2 * t]; a.y = areg[2 * t + 1];
      v2f bb = *(const v2f*)(&sBT[n * 64 + 4 * t + kb]);
      acc = wmma4(a, bb, acc);
    }
    V8U u; u.v = acc;
    float b1 = fc1b[n], w2 = fc2w[n];
#pragma unroll
    for (int v = 0; v < 8; ++v)
      psum[v] += gelu_tanh(u.f[v] + b1) * w2;
  }
  // reduce over the 16 columns (lanes within each half of the wave)
#pragma unroll
  for (int v = 0; v < 8; ++v) {
    float x = psum[v];
    x += __shfl_xor(x, 8, 32);
    x += __shfl_xor(x, 4, 32);
    x += __shfl_xor(x, 2, 32);
    x += __shfl_xor(x, 1, 32);
    psum[v] = x;
  }
  if ((lane & 15) == 0) {
    int roff = (lane >> 4) * 8;
    float b2 = fc2b[0];
#pragma unroll
    for (int v = 0; v < 8; ++v)
      out[(size_t)b * 16384 + s + roff + v] = psum[v] + b2;
  }
}

extern "C" void kernel_launch(void* const* d_in, const int* in_sizes, int n_in,
                              void* d_out, int out_size, void* d_ws, size_t ws_size,
                              hipStream_t stream) {
  (void)in_sizes; (void)n_in; (void)out_size; (void)ws_size;
  const float* x     = (const float*)d_in[0];
  const float* meshx = (const float*)d_in[1];
  const float* meshy = (const float*)d_in[2];
  const float* fc0w  = (const float*)d_in[3];
  const float* fc0b  = (const float*)d_in[4];
  const float* scw1  = (const float*)d_in[5];
  const float* scw2  = (const float*)d_in[6];
  const float* ww    = (const float*)d_in[7];
  const float* wb    = (const float*)d_in[8];
  const float* fc1w  = (const float*)d_in[9];
  const float* fc1b  = (const float*)d_in[10];
  const float* fc2w  = (const float*)d_in[11];
  const float* fc2b  = (const float*)d_in[12];

  float* ws = (float*)d_ws;
  float* H0    = ws;                    // 16,777,216
  float* H1    = ws + 16777216;         // 16,777,216
  float* yhatT = ws + 33554432;         // 8,388,608  (1024 x 32 x 256; rows>=20 junk)
  float* xft   = ws + 41943040;         // 1,638,400  (40 x 20 x 2 x 16 x 64)
  float* oft2  = ws + 43581440;         // 3,932,160  (16 x 64 x 48 x 80)
  float* z     = ws + 47513600;         // 5,242,880  (16 x 64 x 128 x 40)
  float* wprep = ws + 52756480;         // 6,553,600  (40 x 20 x 2 x 64 x 64)
  float* FyT   = ws + 59310080;         // 6,144
  float* Fx2T  = ws + 59316224;         // 20,480
  float* E2    = ws + 59336704;         // 10,240
  float* GyT   = ws + 59346944;         // 5,120

  k_bases<<<164, 256, 0, stream>>>(FyT, Fx2T, E2, GyT);
  k_fc0<<<65536, 256, 0, stream>>>(x, meshx, meshy, fc0w, fc0b, H0);

  for (int l = 0; l < 4; ++l) {
    float* Hin  = (l & 1) ? H1 : H0;
    float* Hout = (l & 1) ? H0 : H1;
    k_prepw<<<25600, 256, 0, stream>>>(scw1, scw2, wprep, l);
    k_conv <<<2048,  256, 0, stream>>>(Hin, Hout, ww, l);
    k_dfty <<<1024,  256, 0, stream>>>(Hin, yhatT, FyT);
    k_dftx <<<1280,  256, 0, stream>>>(yhatT, xft, Fx2T);
    k_cmix <<<400,   256, 0, stream>>>(xft, wprep, oft2);
    k_idftx<<<3072,  256, 0, stream>>>(oft2, z, E2);
    k_idfty<<<1024,  256, 0, stream>>>(z, Hout, GyT, wb, l);
  }
  // after l=3 the activations are in H0
  k_head<<<2048, 256, 0, stream>>>(H0, fc1w, fc1b, fc2w, fc2b, (float*)d_out);
}